// SwavLoss_48137993453687
// MI455X (gfx1250) — compile-verified
//
#include <hip/hip_runtime.h>
#include <hip/hip_bf16.h>

// SwAV loss for MI455X (gfx1250, wave32).
// Shapes: B=1024, K=3000, D=128, LQ=7168, TOT=8, crops {0,1}.
// E (exp matrix) = 8192x3000 f32 in workspace (98MB, fits 192MB global L2).
// Note: 3000 floats = 12000 bytes per row, divisible by 16 -> float4/b128 ok.

typedef __attribute__((ext_vector_type(2))) float v2f;
typedef __attribute__((ext_vector_type(8))) float v8f;

#define KDIM   3000
#define K4DIM  750             // KDIM / 4 (exact)
#define DDIM   128
#define BROWS  1024
#define LQROWS 7168
#define NROWS  8192            // LQ + B
#define INV_EPS 20.0f          // 1/0.05
#define INV_T   10.0f          // 1/0.1

#if defined(__has_builtin)
#  if __has_builtin(__builtin_amdgcn_global_load_async_to_lds_b32) && \
      __has_builtin(__builtin_amdgcn_global_store_async_from_lds_b32) && \
      __has_builtin(__builtin_amdgcn_s_wait_asynccnt)
#    define HAVE_ASYNC_LDS 1
#  endif
#endif

__device__ __forceinline__ float wave_sum(float x) {
    for (int off = 16; off > 0; off >>= 1) x += __shfl_down(x, off, 32);
    return x;
}
__device__ __forceinline__ float wave_max(float x) {
    for (int off = 16; off > 0; off >>= 1) x = fmaxf(x, __shfl_down(x, off, 32));
    return x;
}

// ---------------------------------------------------------------------------
// New queue: nq[r] = r<1024 ? emb_slice[r] : q_old[r-1024]  (7168x128 floats).
// nq = d_out+1 is only 4B-aligned, so this is a b32 copy; on gfx1250 we route
// it through LDS with the async copy engine (ASYNCcnt) to exercise that path.
// grid = 896 blocks x 256 threads, 4 elements/thread (exact, no tail).
// ---------------------------------------------------------------------------
__global__ __launch_bounds__(256) void queue_shift_kernel(
    const float* __restrict__ emb, const float* __restrict__ qold,
    float* __restrict__ nq) {
    const int base = blockIdx.x * 1024;
#if HAVE_ASYNC_LDS
    __shared__ float buf[1024];
    typedef __attribute__((address_space(3))) int lint;
    typedef __attribute__((address_space(1))) int gint;
#pragma unroll
    for (int j = 0; j < 4; ++j) {
        int idx = base + j * 256 + threadIdx.x;
        const float* src = (idx < BROWS * DDIM) ? (emb + idx)
                                                : (qold + (idx - BROWS * DDIM));
        __builtin_amdgcn_global_load_async_to_lds_b32(
            (gint*)src, (lint*)&buf[j * 256 + threadIdx.x], 0, 0);
    }
    __builtin_amdgcn_s_wait_asynccnt(0);
#pragma unroll
    for (int j = 0; j < 4; ++j) {
        int idx = base + j * 256 + threadIdx.x;
        __builtin_amdgcn_global_store_async_from_lds_b32(
            (gint*)(nq + idx), (lint*)&buf[j * 256 + threadIdx.x], 0, 0);
    }
    __builtin_amdgcn_s_wait_asynccnt(0);
#else
#pragma unroll
    for (int j = 0; j < 4; ++j) {
        int idx = base + j * 256 + threadIdx.x;
        nq[idx] = (idx < BROWS * DDIM) ? emb[idx] : qold[idx - BROWS * DDIM];
    }
#endif
}

// ---------------------------------------------------------------------------
// WMMA GEMM:  E[b,k] = exp( (queue_row_b . proto_row_k) / eps ),  b<7168.
// Wave computes a 16x64 C tile via v_wmma_f32_16x16x4_f32 (exact f32 path).
// A 16x4 frag: lane%16 = M row, VGPR j holds K = 2*(lane/16)+j  (2 VGPRs).
// B  4x16 frag: lane%16 = N col, same K striping.
// C/D:          M = v + 8*(lane/16), N = lane%16 (8 VGPRs).
// Also accumulates global sum S of stored exp values.
// ---------------------------------------------------------------------------
__global__ __launch_bounds__(256) void gemm_exp_kernel(
    const float* __restrict__ q,      // 7168x128 (queue[i])
    const float* __restrict__ w,      // 3000x128 (proto_weight)
    float* __restrict__ E,            // 8192x3000
    float* __restrict__ S) {
    const int lane = threadIdx.x & 31;
    const int wgid = blockIdx.x * 8 + (threadIdx.x >> 5);
    const int m_tile = wgid % 448;                 // 7168/16 M tiles
    const int ngrp   = wgid / 448;                 // groups of 4 N-tiles (64 cols)
    if (ngrp >= 47) return;                        // ceil(188/4) = 47
    const int m0 = m_tile * 16;
    const int n0 = ngrp * 64;
    const int lm = lane & 15;
    const int lh = lane >> 4;

    const float* arow = q + (m0 + lm) * DDIM + 2 * lh;
    const float* brow[4];
    int ncol[4];
#pragma unroll
    for (int t = 0; t < 4; ++t) {
        int n = n0 + t * 16 + lm;
        ncol[t] = n;
        brow[t] = w + (n < KDIM ? n : (KDIM - 1)) * DDIM + 2 * lh;
    }

    v8f acc[4] = {v8f{}, v8f{}, v8f{}, v8f{}};
    for (int k = 0; k < DDIM; k += 4) {
        v2f a = *(const v2f*)(arow + k);
#pragma unroll
        for (int t = 0; t < 4; ++t) {
            v2f b = *(const v2f*)(brow[t] + k);
            acc[t] = __builtin_amdgcn_wmma_f32_16x16x4_f32(
                false, a, false, b, (short)0, acc[t], false, false);
        }
    }

    float local = 0.0f;
#pragma unroll
    for (int t = 0; t < 4; ++t) {
        if (ncol[t] < KDIM) {
#pragma unroll
            for (int v = 0; v < 8; ++v) {
                int row = m0 + v + 8 * lh;
                float e = __expf(acc[t][v] * INV_EPS);
                E[(size_t)row * KDIM + ncol[t]] = e;
                local += e;
            }
        }
    }
    local = wave_sum(local);
    if (lane == 0) atomicAdd(S, local);
}

// ---------------------------------------------------------------------------
// Bottom 1024 rows of E from the crop's output slice (float4 / b128 path),
// plus S contribution.  grid = 3000 blocks x 256 threads (768000 float4s).
// ---------------------------------------------------------------------------
__global__ __launch_bounds__(256) void exp_rows_kernel(
    const float4* __restrict__ src4,   // output slice, 1024x3000 as float4
    float4* __restrict__ dst4,         // E + 7168*3000, as float4
    float* __restrict__ S) {
    int idx = blockIdx.x * 256 + threadIdx.x;        // < 768000
    float local = 0.0f;
    if (idx < BROWS * K4DIM) {
        float4 x = src4[idx];
        float4 e;
        e.x = __expf(x.x * INV_EPS);
        e.y = __expf(x.y * INV_EPS);
        e.z = __expf(x.z * INV_EPS);
        e.w = __expf(x.w * INV_EPS);
        dst4[idx] = e;
        local = e.x + e.y + e.z + e.w;
    }
    local = wave_sum(local);
    if ((threadIdx.x & 31) == 0) atomicAdd(S, local);
}

__global__ void init_v_kernel(const float* __restrict__ S, float* __restrict__ v) {
    int b = blockIdx.x * 256 + threadIdx.x;
    if (b < NROWS) v[b] = 1.0f / S[0];
}

// colacc[k] += sum_{rows in 32-row chunk} v[b] * E[b,k]
// thread handles 4 consecutive columns (b128 loads).  grid = (3, 256).
__global__ __launch_bounds__(256) void colsum_kernel(
    const float* __restrict__ E, const float* __restrict__ v,
    float* __restrict__ colacc) {
    int c4 = blockIdx.x * 256 + threadIdx.x;
    if (c4 >= K4DIM) return;
    int r0 = blockIdx.y * 32;
    float4 acc = make_float4(0.f, 0.f, 0.f, 0.f);
    for (int r = r0; r < r0 + 32; ++r) {
        float4 e = *(const float4*)(E + (size_t)r * KDIM + c4 * 4);
        float vb = v[r];
        acc.x += vb * e.x; acc.y += vb * e.y;
        acc.z += vb * e.z; acc.w += vb * e.w;
    }
    atomicAdd(&colacc[c4 * 4 + 0], acc.x);
    atomicAdd(&colacc[c4 * 4 + 1], acc.y);
    atomicAdd(&colacc[c4 * 4 + 2], acc.z);
    atomicAdd(&colacc[c4 * 4 + 3], acc.w);
}

__global__ void u_update_kernel(const float* __restrict__ colacc,
                                float* __restrict__ u) {
    int k = blockIdx.x * 256 + threadIdx.x;
    if (k < KDIM) u[k] = 1.0f / ((float)KDIM * colacc[k]);
}

// v[b] = 1 / (8192 * sum_k u[k]*E[b,k]),  one wave per row, float4 loads
__global__ __launch_bounds__(256) void rowsum_v_kernel(
    const float* __restrict__ E, const float* __restrict__ u,
    float* __restrict__ v) {
    int row = blockIdx.x * 8 + (threadIdx.x >> 5);
    int lane = threadIdx.x & 31;
    const float4* er4 = (const float4*)(E + (size_t)row * KDIM);
    const float4* u4  = (const float4*)u;
    float acc = 0.0f;
    for (int k = lane; k < K4DIM; k += 32) {
        float4 e = er4[k];
        float4 uu = u4[k];
        acc += uu.x * e.x + uu.y * e.y + uu.z * e.z + uu.w * e.w;
    }
    acc = wave_sum(acc);
    if (lane == 0) v[row] = 1.0f / ((float)NROWS * acc);
}

// lse[row] = logsumexp(output[row,:] / T),  one wave per row (8192 rows)
__global__ __launch_bounds__(256) void lse_kernel(
    const float* __restrict__ out, float* __restrict__ lse) {
    int row = blockIdx.x * 8 + (threadIdx.x >> 5);
    int lane = threadIdx.x & 31;
    const float4* xr4 = (const float4*)(out + (size_t)row * KDIM);
    float m = -3.0e38f;
    for (int k = lane; k < K4DIM; k += 32) {
        float4 x = xr4[k];
        m = fmaxf(m, fmaxf(fmaxf(x.x, x.y), fmaxf(x.z, x.w)) * INV_T);
    }
    m = wave_max(m);
    m = __shfl(m, 0, 32);
    float s = 0.0f;
    for (int k = lane; k < K4DIM; k += 32) {
        float4 x = xr4[k];
        s += __expf(x.x * INV_T - m) + __expf(x.y * INV_T - m)
           + __expf(x.z * INV_T - m) + __expf(x.w * INV_T - m);
    }
    s = wave_sum(s);
    if (lane == 0) lse[row] = __logf(s) + m;
}

// One wave per (view, row): acc += 8192*v[7168+r]*(sum_k u*E*x)*10 - lse
__global__ __launch_bounds__(256) void loss_kernel(
    const float* __restrict__ E, const float* __restrict__ u,
    const float* __restrict__ v, const float* __restrict__ output,
    const float* __restrict__ lse, float* __restrict__ acc, int crop) {
    int wid = blockIdx.x * 8 + (threadIdx.x >> 5);   // 0..7167
    int lane = threadIdx.x & 31;
    int r  = wid & 1023;
    int ve = wid >> 10;                              // 0..6
    int vi = ve + (ve >= crop ? 1 : 0);              // skip crop view
    const float4* er4 = (const float4*)(E + (size_t)(LQROWS + r) * KDIM);
    const float4* xr4 = (const float4*)(output + (size_t)(vi * BROWS + r) * KDIM);
    const float4* u4  = (const float4*)u;
    float d = 0.0f;
    for (int k = lane; k < K4DIM; k += 32) {
        float4 e = er4[k];
        float4 x = xr4[k];
        float4 uu = u4[k];
        d += uu.x * e.x * x.x + uu.y * e.y * x.y
           + uu.z * e.z * x.z + uu.w * e.w * x.w;
    }
    d = wave_sum(d);
    if (lane == 0) {
        float term = (float)NROWS * v[LQROWS + r] * (d * INV_T)
                     - lse[vi * BROWS + r];
        atomicAdd(acc, term);
    }
}

__global__ void finalize_kernel(const float* __restrict__ acc,
                                float* __restrict__ out) {
    out[0] = -acc[0] * (1.0f / 14336.0f);   // 1/(2 crops * 7 views * 1024 rows)
}

// ---------------------------------------------------------------------------
extern "C" void kernel_launch(void* const* d_in, const int* in_sizes, int n_in,
                              void* d_out, int out_size, void* d_ws, size_t ws_size,
                              hipStream_t stream) {
    const float* output    = (const float*)d_in[1];   // 8192x3000
    const float* embedding = (const float*)d_in[2];   // 8192x128
    const float* queue     = (const float*)d_in[3];   // 2x7168x128
    const float* proto     = (const float*)d_in[4];   // 3000x128
    float* out = (float*)d_out;                       // [loss | new_queue 2x7168x128]

    float* E      = (float*)d_ws;                     // 8192*3000
    float* colacc = E + (size_t)NROWS * KDIM;         // 3000
    float* u      = colacc + KDIM;                    // 3000
    float* v      = u + KDIM;                         // 8192
    float* lse    = v + NROWS;                        // 8192
    float* S      = lse + NROWS;                      // 1
    float* lacc   = S + 1;                            // 1

    (void)hipMemsetAsync(lacc, 0, sizeof(float), stream);
    lse_kernel<<<1024, 256, 0, stream>>>(output, lse);

    for (int i = 0; i < 2; ++i) {
        queue_shift_kernel<<<896, 256, 0, stream>>>(
            embedding + (size_t)i * BROWS * DDIM,
            queue + (size_t)i * LQROWS * DDIM,
            out + 1 + (size_t)i * LQROWS * DDIM);

        (void)hipMemsetAsync(S, 0, sizeof(float), stream);
        gemm_exp_kernel<<<2632, 256, 0, stream>>>(
            queue + (size_t)i * LQROWS * DDIM, proto, E, S);
        exp_rows_kernel<<<3000, 256, 0, stream>>>(
            (const float4*)(output + (size_t)i * BROWS * KDIM),
            (float4*)(E + (size_t)LQROWS * KDIM), S);
        init_v_kernel<<<32, 256, 0, stream>>>(S, v);

        for (int it = 0; it < 3; ++it) {
            (void)hipMemsetAsync(colacc, 0, KDIM * sizeof(float), stream);
            colsum_kernel<<<dim3(3, 256), 256, 0, stream>>>(E, v, colacc);
            u_update_kernel<<<12, 256, 0, stream>>>(colacc, u);
            rowsum_v_kernel<<<1024, 256, 0, stream>>>(E, u, v);
        }

        loss_kernel<<<896, 256, 0, stream>>>(E, u, v, output, lse, lacc, i);
    }

    finalize_kernel<<<1, 1, 0, stream>>>(lacc, out);
}